// Head_19610820674177
// MI455X (gfx1250) — compile-verified
//
#include <hip/hip_runtime.h>
#include <hip/hip_bf16.h>

typedef __attribute__((ext_vector_type(2))) float v2f;
typedef __attribute__((ext_vector_type(8))) float v8f;

#define BATCH 4
#define SEQ   2048
#define DM    1024
#define DH    64
#define ROWS  (BATCH * SEQ)        // 8192
#define QT_PER_B (SEQ / 16)        // 128
#define NQT   (BATCH * QT_PER_B)   // 512

static __device__ __forceinline__ v8f wmma4(v2f a, v2f b, v8f c) {
  // D = A(16x4 f32) * B(4x16 f32) + C(16x16 f32), native fp32 WMMA
  return __builtin_amdgcn_wmma_f32_16x16x4_f32(false, a, false, b, (short)0, c,
                                               false, false);
}

// ---------------------------------------------------------------------------
// Kernel 1: QKV projection.  One block = one 16-row tile of x.
// 12 waves: wave w -> matrix (w>>2 in {Q,K,V}), col tile (w&3)*16.
// x tile staged in LDS transposed [k][row] (bank-conflict-free A reads).
// ---------------------------------------------------------------------------
__global__ __launch_bounds__(384) void qkv_proj_kernel(
    const float* __restrict__ x, const float* __restrict__ Wq,
    const float* __restrict__ Wk, const float* __restrict__ Wv,
    float* __restrict__ qkv /* [3][ROWS][DH] */) {
  __shared__ float xs[DM * 16];  // xs[k*16 + row], 64 KB of the 320 KB WGP LDS
  const int tid  = threadIdx.x;
  const int lane = tid & 31;
  const int wv   = tid >> 5;
  const int row0 = blockIdx.x * 16;

  // Cooperative transpose-load of the x tile into LDS (float4 global loads).
  const float4* xg = (const float4*)(x + (size_t)row0 * DM);
  for (int i = tid; i < 16 * (DM / 4); i += 384) {
    const int r = i / (DM / 4);
    const int c = i % (DM / 4);
    float4 v = xg[(size_t)r * (DM / 4) + c];
    xs[(4 * c + 0) * 16 + r] = v.x;
    xs[(4 * c + 1) * 16 + r] = v.y;
    xs[(4 * c + 2) * 16 + r] = v.z;
    xs[(4 * c + 3) * 16 + r] = v.w;
  }
  __syncthreads();

  const int m  = wv >> 2;           // 0=Q 1=K 2=V
  const int n0 = (wv & 3) * 16;
  const float* W = (m == 0) ? Wq : (m == 1) ? Wk : Wv;
  float* out = qkv + (size_t)m * ROWS * DH;
  const int h = lane >> 4, ln = lane & 15;

  v8f acc = {};
  for (int k = 0; k < DM; k += 4) {
    v2f a;  // A[m=ln][k + 2h + j]
    a[0] = xs[(k + 2 * h + 0) * 16 + ln];
    a[1] = xs[(k + 2 * h + 1) * 16 + ln];
    v2f b;  // B[k'+h][n=ln], B[k'+2+h][n=ln]
    b[0] = W[(size_t)(k + h) * DH + n0 + ln];
    b[1] = W[(size_t)(k + 2 + h) * DH + n0 + ln];
    acc = wmma4(a, b, acc);
  }
#pragma unroll
  for (int r = 0; r < 8; ++r)
    out[(size_t)(row0 + r + 8 * h) * DH + n0 + ln] = acc[r];
}

// ---------------------------------------------------------------------------
// Kernel 2: causal flash attention, fp32 WMMA.
// 512 single-wave workgroups: up to 32 WGs per WGP stack multiple waves per
// SIMD for latency hiding, barrier ops degrade to S_NOP, and the causal work
// skew spreads across 512 schedulable units.
// ---------------------------------------------------------------------------
__global__ __launch_bounds__(32) void attn_kernel(
    const float* __restrict__ qkv, float* __restrict__ out) {
  __shared__ float pbuf[16][17];  // P staging, padded vs bank conflicts
  const int lane = threadIdx.x & 31;
  const int qt   = blockIdx.x;
  const int b    = qt / QT_PER_B;
  const int r0   = (qt % QT_PER_B) * 16;

  const float* Q = qkv + (size_t)b * SEQ * DH;
  const float* K = qkv + (size_t)ROWS * DH + (size_t)b * SEQ * DH;
  const float* V = qkv + 2 * (size_t)ROWS * DH + (size_t)b * SEQ * DH;
  const int h = lane >> 4, ln = lane & 15;

  // Preload Q tile as sixteen 16x4 A-chunks, folding in 1/sqrt(DH).
  const float scale = 0.125f;
  v2f q[16];
#pragma unroll
  for (int j = 0; j < 16; ++j) {
    const float* qp = &Q[(size_t)(r0 + ln) * DH + 4 * j + 2 * h];
    q[j][0] = qp[0] * scale;
    q[j][1] = qp[1] * scale;
  }

  v8f o0 = {}, o1 = {}, o2 = {}, o3 = {};
  float mrow[8], lrow[8];
#pragma unroll
  for (int r = 0; r < 8; ++r) { mrow[r] = -1e30f; lrow[r] = 0.0f; }

  for (int kb = 0; kb <= r0; kb += 16) {
    // Prefetch next K block into cache (global_prefetch_b8).
    __builtin_prefetch(&K[(size_t)(kb + 16 + ln) * DH], 0, 1);

    // ---- S = (Q*scale) . K^T : 16 WMMA over d_head ----
    v8f s = {};
#pragma unroll
    for (int j = 0; j < 16; ++j) {
      float4 kv = *(const float4*)&K[(size_t)(kb + ln) * DH + 4 * j];
      v2f bv;  // B[k'][n]: k' = 4j + {h, 2+h}, n = key lane
      bv[0] = h ? kv.y : kv.x;
      bv[1] = h ? kv.w : kv.z;
      s = wmma4(q[j], bv, s);
    }

    // ---- causal mask on the diagonal block ----
    if (kb == r0) {
#pragma unroll
      for (int r = 0; r < 8; ++r) {
        const int row = r + 8 * h;
        if (ln > row) s[r] = -1e30f;
      }
    }

    // ---- online softmax (row stats live on 16-lane halves) ----
    float fsc[8];
#pragma unroll
    for (int r = 0; r < 8; ++r) {
      float mx = s[r];
      mx = fmaxf(mx, __shfl_xor(mx, 1, 16));
      mx = fmaxf(mx, __shfl_xor(mx, 2, 16));
      mx = fmaxf(mx, __shfl_xor(mx, 4, 16));
      mx = fmaxf(mx, __shfl_xor(mx, 8, 16));
      const float mnew = fmaxf(mrow[r], mx);
      fsc[r] = __expf(mrow[r] - mnew);
      const float p = __expf(s[r] - mnew);
      s[r] = p;
      float sum = p;
      sum += __shfl_xor(sum, 1, 16);
      sum += __shfl_xor(sum, 2, 16);
      sum += __shfl_xor(sum, 4, 16);
      sum += __shfl_xor(sum, 8, 16);
      lrow[r] = lrow[r] * fsc[r] + sum;
      mrow[r] = mnew;
    }
#pragma unroll
    for (int r = 0; r < 8; ++r) {
      o0[r] *= fsc[r]; o1[r] *= fsc[r]; o2[r] *= fsc[r]; o3[r] *= fsc[r];
    }

    // ---- stage P (C-layout -> A-layout) through LDS, same wave only ----
#pragma unroll
    for (int r = 0; r < 8; ++r)
      pbuf[r + 8 * h][ln] = s[r];
    asm volatile("s_wait_dscnt 0" ::: "memory");  // CDNA5 split DS counter

    // ---- O += P . V : 4 chunks x 4 col tiles ----
#pragma unroll
    for (int c = 0; c < 4; ++c) {
      v2f a;  // A[m=ln][k'=4c+2h+j]
      a[0] = pbuf[ln][4 * c + 2 * h + 0];
      a[1] = pbuf[ln][4 * c + 2 * h + 1];
      const float* v0 = &V[(size_t)(kb + 4 * c + h) * DH];
      const float* v1 = &V[(size_t)(kb + 4 * c + 2 + h) * DH];
      v2f b0v, b1v, b2v, b3v;
      b0v[0] = v0[ 0 + ln]; b0v[1] = v1[ 0 + ln];
      b1v[0] = v0[16 + ln]; b1v[1] = v1[16 + ln];
      b2v[0] = v0[32 + ln]; b2v[1] = v1[32 + ln];
      b3v[0] = v0[48 + ln]; b3v[1] = v1[48 + ln];
      o0 = wmma4(a, b0v, o0);
      o1 = wmma4(a, b1v, o1);
      o2 = wmma4(a, b2v, o2);
      o3 = wmma4(a, b3v, o3);
    }
  }

  // ---- normalize and store ----
  float* op = out + (size_t)(b * SEQ + r0) * DH;
#pragma unroll
  for (int r = 0; r < 8; ++r) {
    const float inv = 1.0f / lrow[r];
    const int row = r + 8 * h;
    op[(size_t)row * DH +  0 + ln] = o0[r] * inv;
    op[(size_t)row * DH + 16 + ln] = o1[r] * inv;
    op[(size_t)row * DH + 32 + ln] = o2[r] * inv;
    op[(size_t)row * DH + 48 + ln] = o3[r] * inv;
  }
}

extern "C" void kernel_launch(void* const* d_in, const int* in_sizes, int n_in,
                              void* d_out, int out_size, void* d_ws,
                              size_t ws_size, hipStream_t stream) {
  const float* x  = (const float*)d_in[0];
  const float* Wq = (const float*)d_in[1];
  const float* Wk = (const float*)d_in[2];
  const float* Wv = (const float*)d_in[3];
  float* qkv = (float*)d_ws;  // 3 * 8192 * 64 * 4 = 6 MB scratch

  qkv_proj_kernel<<<ROWS / 16, 384, 0, stream>>>(x, Wq, Wk, Wv, qkv);
  attn_kernel<<<NQT, 32, 0, stream>>>(qkv, (float*)d_out);
}